// FourierModelTestNU_72705206387341
// MI455X (gfx1250) — compile-verified
//
#include <hip/hip_runtime.h>
#include <hip/hip_bf16.h>
#include <math.h>

// Problem sizes (fixed by the reference).
#define NN   128
#define TT   1024
#define SS   1024
#define H1S  512
#define H2S  128
#define OUTS 8
#define TWO_PI_F 6.283185307179586f

typedef __attribute__((ext_vector_type(2))) float v2f;
typedef __attribute__((ext_vector_type(8))) float v8f;

// ---------------------------------------------------------------------------
// Kernel 1: per-sequence non-uniform Fourier magnitudes.
//   freq[n,s] = | sum_t exp(-i*2*pi*s*day[n,t]) * x[n,t] |
// One block per n (256 threads = 8 wave32). Each thread owns 4 consecutive
// s values; base angle via one __sincosf per (thread,t), remaining 3 via
// Chebyshev/rotation recurrence with per-t cos/sin staged in LDS.
// ---------------------------------------------------------------------------
__global__ __launch_bounds__(256) void nufft_mag_kernel(
    const float* __restrict__ inp,   // [NN][TT][3]
    float* __restrict__ freq) {      // [NN][SS]
  __shared__ float xs[TT];   // x[n,t]
  __shared__ float th[TT];   // theta_t = -2*pi*day[n,t]
  __shared__ float ct[TT];   // cos(theta_t)
  __shared__ float st[TT];   // sin(theta_t)

  const int n   = blockIdx.x;
  const int tid = threadIdx.x;

  for (int t = tid; t < TT; t += 256) {
    const float x = inp[(n * TT + t) * 3 + 0];
    const float d = inp[(n * TT + t) * 3 + 2];
    const float theta = -TWO_PI_F * d;
    float s, c;
    __sincosf(theta, &s, &c);
    xs[t] = x; th[t] = theta; ct[t] = c; st[t] = s;
  }
  __syncthreads();

  const float s0 = (float)(tid * 4);
  float re0 = 0.f, re1 = 0.f, re2 = 0.f, re3 = 0.f;
  float im0 = 0.f, im1 = 0.f, im2 = 0.f, im3 = 0.f;

  for (int t = 0; t < TT; ++t) {
    const float xv    = xs[t];
    const float theta = th[t];
    const float cr    = ct[t];
    const float sr    = st[t];

    float sb, cb;                       // sin/cos of s0*theta
    __sincosf(s0 * theta, &sb, &cb);

    re0 = fmaf(cb, xv, re0);  im0 = fmaf(sb, xv, im0);
    const float c1 = fmaf(cb, cr, -sb * sr);
    const float s1 = fmaf(sb, cr,  cb * sr);
    re1 = fmaf(c1, xv, re1);  im1 = fmaf(s1, xv, im1);
    const float c2 = fmaf(c1, cr, -s1 * sr);
    const float s2 = fmaf(s1, cr,  c1 * sr);
    re2 = fmaf(c2, xv, re2);  im2 = fmaf(s2, xv, im2);
    const float c3 = fmaf(c2, cr, -s2 * sr);
    const float s3 = fmaf(s2, cr,  c2 * sr);
    re3 = fmaf(c3, xv, re3);  im3 = fmaf(s3, xv, im3);
  }

  float* fr = freq + n * SS + tid * 4;
  fr[0] = sqrtf(fmaf(re0, re0, im0 * im0));
  fr[1] = sqrtf(fmaf(re1, re1, im1 * im1));
  fr[2] = sqrtf(fmaf(re2, re2, im2 * im2));
  fr[3] = sqrtf(fmaf(re3, re3, im3 * im3));
}

// ---------------------------------------------------------------------------
// Kernel 2/3: fused GEMM + bias + optional sigmoid using fp32 WMMA.
//   Out[m,n] = act( sum_k A[m,k] * W[n,k] + bias[n] )
// One wave32 per 16x16 C tile, K stepped by 4 via V_WMMA_F32_16X16X4_F32.
// A-frag (16x4): lane<16 -> K={kk,kk+1}, lane>=16 -> K={kk+2,kk+3}, row=lane&15.
// B-frag (4x16): same K split, col=lane&15 (B = W^T so rows of W are columns).
// C/D: VGPR v -> row m0 + v + 8*(lane>=16), col n0 + (lane&15).
// ---------------------------------------------------------------------------
__global__ __launch_bounds__(128) void gemm_bias_act_kernel(
    const float* __restrict__ A,     // M x K row-major
    const float* __restrict__ W,     // Nn x K row-major
    const float* __restrict__ bias,  // Nn
    float* __restrict__ Out,         // M x Nn
    int M, int Nn, int K, int apply_sigmoid) {
  const int lane   = threadIdx.x & 31;
  const int wave   = threadIdx.x >> 5;
  const int ntiles = Nn >> 4;
  const int tile   = blockIdx.x * 4 + wave;
  const int mt = tile / ntiles;
  const int nt = tile - mt * ntiles;
  const int m0 = mt << 4;
  const int n0 = nt << 4;

  const int half = lane >> 4;       // 0: K pair {0,1}; 1: K pair {2,3}
  const int lm   = lane & 15;

  const float* __restrict__ arow = A + (size_t)(m0 + lm) * K + 2 * half;
  const float* __restrict__ brow = W + (size_t)(n0 + lm) * K + 2 * half;

  v8f acc = {};
#pragma unroll 4
  for (int kk = 0; kk < K; kk += 4) {
    const v2f a = *(const v2f*)(arow + kk);
    const v2f b = *(const v2f*)(brow + kk);
    acc = __builtin_amdgcn_wmma_f32_16x16x4_f32(
        /*neg_a=*/false, a, /*neg_b=*/false, b,
        /*c_mod=*/(short)0, acc, /*reuse_a=*/false, /*reuse_b=*/false);
  }

  const float bv = bias[n0 + lm];
#pragma unroll
  for (int v = 0; v < 8; ++v) {
    float val = acc[v] + bv;
    if (apply_sigmoid) val = 1.0f / (1.0f + __expf(-val));
    Out[(size_t)(m0 + v + 8 * half) * Nn + (n0 + lm)] = val;
  }
}

// ---------------------------------------------------------------------------
// Kernel 4: tiny head layer, out[n,o] = dot(H2[n,:], W3[o,:]) + b3[o].
// 1024 independent length-128 dots -> plain VALU.
// ---------------------------------------------------------------------------
__global__ __launch_bounds__(256) void head_kernel(
    const float* __restrict__ H2,    // NN x H2S
    const float* __restrict__ W3,    // OUTS x H2S
    const float* __restrict__ b3,    // OUTS
    float* __restrict__ out) {       // NN x OUTS
  const int idx = blockIdx.x * 256 + threadIdx.x;   // 0..NN*OUTS-1
  const int n = idx >> 3;
  const int o = idx & 7;
  const float* __restrict__ h = H2 + n * H2S;
  const float* __restrict__ w = W3 + o * H2S;
  float acc = 0.f;
#pragma unroll 8
  for (int k = 0; k < H2S; ++k) acc = fmaf(h[k], w[k], acc);
  out[idx] = acc + b3[o];
}

// ---------------------------------------------------------------------------
extern "C" void kernel_launch(void* const* d_in, const int* in_sizes, int n_in,
                              void* d_out, int out_size, void* d_ws, size_t ws_size,
                              hipStream_t stream) {
  const float* inp = (const float*)d_in[0];   // [128,1024,3]
  const float* W1  = (const float*)d_in[1];   // [512,1024]
  const float* b1  = (const float*)d_in[2];   // [512]
  const float* W2  = (const float*)d_in[3];   // [128,512]
  const float* b2  = (const float*)d_in[4];   // [128]
  const float* W3  = (const float*)d_in[5];   // [8,128]
  const float* b3  = (const float*)d_in[6];   // [8]
  float* out = (float*)d_out;

  float* F  = (float*)d_ws;          // 128*1024 f32 = 512 KB
  float* H1 = F  + NN * SS;          // 128*512  f32 = 256 KB
  float* H2 = H1 + NN * H1S;         // 128*128  f32 =  64 KB

  // Phase 1: Fourier magnitudes, one block per sequence.
  nufft_mag_kernel<<<NN, 256, 0, stream>>>(inp, F);

  // Phase 2: H1 = sigmoid(F @ W1^T + b1). Tiles: (128/16)*(512/16)=256 waves.
  gemm_bias_act_kernel<<<(8 * 32) / 4, 128, 0, stream>>>(
      F, W1, b1, H1, NN, H1S, SS, 1);

  // Phase 3: H2 = sigmoid(H1 @ W2^T + b2). Tiles: 8*8=64 waves.
  gemm_bias_act_kernel<<<(8 * 8) / 4, 128, 0, stream>>>(
      H1, W2, b2, H2, NN, H2S, H1S, 1);

  // Phase 4: out = H2 @ W3^T + b3.
  head_kernel<<<(NN * OUTS) / 256, 256, 0, stream>>>(H2, W3, b3, out);
}